// BicycleModel_49563922596139
// MI455X (gfx1250) — compile-verified
//
#include <hip/hip_runtime.h>

// Bicycle-model rollout for MI455X (gfx1250, wave32).
// B=65536 trajectories, 256 time steps, strictly sequential in t, parallel in B.
// HBM-bound (~385 MB total traffic => ~16.5 us floor at 23.3 TB/s).
//
// CDNA5-specific paths:
//  - Tensor Data Mover: TENSOR_LOAD_TO_LDS (inline asm, 2-group D#) DMA's each
//    256-row x 64B control tile straight into LDS; hardware pad_interval/amount
//    reproduces the bank-conflict-free pitch-17 layout. Tracked by TENSORcnt
//    (s_wait_tensorcnt). One leader wave issues 2 DMAs per block per tile.
//  - Double-buffered LDS: DMA of tile k+1 overlaps compute of tile k.
//  - global_prefetch_b8 via __builtin_prefetch.
//  - float4 (global_store_b128) output stores; 192 MB L2 write-combines the
//    64B-per-tile fragments into full 128B lines before HBM writeback.

namespace {
constexpr int   N_STEPS   = 256;
constexpr int   TILE      = 16;            // time steps per LDS tile
constexpr int   BLOCK     = 256;           // threads per block = 8 wave32
constexpr int   LDS_PITCH = TILE + 1;      // 17 dwords (odd => conflict-free)
constexpr int   NTILES    = N_STEPS / TILE;
constexpr float DT        = 0.05f;
constexpr float INV_WB    = 1.0f / 2.7f;   // 1 / WHEELBASE
constexpr float MAX_STEER = 0.5235987755982988f;  // deg2rad(30)
constexpr float MAX_SPEED = 100.0f;
}

typedef __attribute__((ext_vector_type(4))) unsigned int uint32x4;
typedef __attribute__((ext_vector_type(8))) unsigned int uint32x8;

// Issue one TDM descriptor: 2D tile, 256 rows x TILE dwords, row stride
// N_STEPS dwords in global, pitch LDS_PITCH dwords in LDS (via pad fields).
// D# bit layout per CDNA5 ISA.
__device__ __forceinline__ void tdm_issue_tile(const float* gsrc, const float* ldst)
{
    const unsigned long long ga = (unsigned long long)(uintptr_t)gsrc;

    uint32x4 g0;
    g0.x = 1u;                                        // count=1, user descriptor
    g0.y = (unsigned int)(uintptr_t)ldst;             // lds_addr (low 32 = LDS offs)
    g0.z = (unsigned int)ga;                          // global_addr[31:0]
    g0.w = ((unsigned int)(ga >> 32) & 0x01FFFFFFu)   // global_addr[56:32]
         | 0x80000000u;                               // type=2 ("image")

    uint32x8 g1;
    g1[0] = (2u << 16)        // data_size = 4 bytes
          | (1u << 20)        // pad_enable
          | (3u << 22)        // pad_interval: every 16 dwords (64B)
          | (0u << 25);       // pad_amount: 1 dword  => pitch 17 dwords
    g1[1] = 0u;               // atomic_barrier_addr=0; tensor_dim0[15:0]=0
    g1[2] = 0x4000u;          // tensor_dim0 = 0x40000000 (huge, no OOB clip)
    g1[3] = 0x4000u                      // tensor_dim1 = 0x40000000
          | ((unsigned int)TILE << 16);  // tile_dim0 = 16 elements (64B row)
    g1[4] = (unsigned int)BLOCK;         // tile_dim1 = 256 rows; tile_dim2 = 0
    g1[5] = (unsigned int)N_STEPS;       // tensor_dim0_stride = 256 elements
    g1[6] = 0u;                          // stride hi; tensor_dim1_stride lo
    g1[7] = 0u;

    asm volatile("tensor_load_to_lds %0, %1" :: "s"(g0), "s"(g1) : "memory");
}

__device__ __forceinline__ void wait_tensorcnt0() {
#if __has_builtin(__builtin_amdgcn_s_wait_tensorcnt)
    __builtin_amdgcn_s_wait_tensorcnt(0);
#else
    asm volatile("s_wait_tensorcnt 0x0" ::: "memory");
#endif
}

__global__ __launch_bounds__(BLOCK) void bicycle_rollout_kernel(
    const float* __restrict__ start_x,
    const float* __restrict__ start_y,
    const float* __restrict__ start_yaw,
    const float* __restrict__ start_speed,
    const float* __restrict__ accel,
    const float* __restrict__ steering,
    float* __restrict__ out,
    size_t nb /* = B * N_STEPS, stride between output arrays */)
{
    __shared__ float sAcc[2][BLOCK * LDS_PITCH];   // 2 x 17.0 KB
    __shared__ float sSt [2][BLOCK * LDS_PITCH];   // 2 x 17.0 KB  (69.6 KB total)

    const int  tid    = threadIdx.x;
    const int  b      = blockIdx.x * BLOCK + tid;
    const bool leader = (tid < 32);                 // wave 0 drives the TDM

    float x   = start_x[b];
    float y   = start_y[b];
    float yaw = start_yaw[b];
    float sp  = start_speed[b];

    float* __restrict__ ox  = out;
    float* __restrict__ oy  = out + nb;
    float* __restrict__ oyw = out + 2 * nb;
    float* __restrict__ osp = out + 3 * nb;

    const size_t rowBase   = (size_t)b * N_STEPS;
    const size_t blockBase = (size_t)(blockIdx.x * BLOCK) * N_STEPS;

    // Prime the pipeline: DMA tile 0 into buffer 0.
    if (leader) {
        tdm_issue_tile(accel    + blockBase, &sAcc[0][0]);
        tdm_issue_tile(steering + blockBase, &sSt [0][0]);
    }

    for (int tile = 0; tile < NTILES; ++tile) {
        const int tbase = tile * TILE;
        const int cur   = tile & 1;

        // Warm L2 for the tile after next (gfx1250: global_prefetch_b8).
        if (tile + 2 < NTILES) {
            __builtin_prefetch(accel    + rowBase + tbase + 2 * TILE, 0, 0);
            __builtin_prefetch(steering + rowBase + tbase + 2 * TILE, 0, 0);
        }

        // Leader drains its TDM queue (tile `tile` is the only outstanding op);
        // the barrier then (a) publishes the landed tile to all waves and
        // (b) guarantees everyone is done reading buffer (tile-1) before the
        // leader overwrites it with DMA(tile+1).
        if (leader) wait_tensorcnt0();
        __syncthreads();

        if (leader && (tile + 1 < NTILES)) {
            const size_t gnext = blockBase + (size_t)(tbase + TILE);
            tdm_issue_tile(accel    + gnext, &sAcc[cur ^ 1][0]);
            tdm_issue_tile(steering + gnext, &sSt [cur ^ 1][0]);
        }

        const float* __restrict__ aRow = &sAcc[cur][tid * LDS_PITCH];
        const float* __restrict__ sRow = &sSt [cur][tid * LDS_PITCH];

        // --- 16 steps, emitted as 4 groups of 4 with float4 (b128) stores.
        #pragma unroll
        for (int j = 0; j < TILE / 4; ++j) {
            float4 vx, vy, vyw, vsp;
            float* px  = reinterpret_cast<float*>(&vx);
            float* py  = reinterpret_cast<float*>(&vy);
            float* pyw = reinterpret_cast<float*>(&vyw);
            float* psp = reinterpret_cast<float*>(&vsp);

            #pragma unroll
            for (int t = 0; t < 4; ++t) {
                // Record state at time (tbase + 4j + t) BEFORE applying its
                // control, matching the reference (out[0] = start state).
                px[t]  = x;
                py[t]  = y;
                pyw[t] = yaw;
                psp[t] = sp;

                const int   tt = j * 4 + t;
                const float a  = aRow[tt];
                const float st = sRow[tt];

                const float fr    = sp * 0.1f + 0.01f * sp * sp;
                const float spn   = fminf(fmaxf(sp + DT * (a - fr), 0.0f), MAX_SPEED);
                const float steer = fminf(fmaxf(st, -MAX_STEER), MAX_STEER);
                const float ang   = sp * __tanf(steer) * INV_WB;   // old speed
                float sy, cy;
                __sincosf(yaw, &sy, &cy);                          // old yaw
                x   += sp * cy * DT;
                y   += sp * sy * DT;
                yaw += ang * DT;
                sp   = spn;
            }

            const size_t o = rowBase + (size_t)(tbase + j * 4);    // 16B aligned
            *reinterpret_cast<float4*>(ox  + o) = vx;
            *reinterpret_cast<float4*>(oy  + o) = vy;
            *reinterpret_cast<float4*>(oyw + o) = vyw;
            *reinterpret_cast<float4*>(osp + o) = vsp;
        }
    }
}

extern "C" void kernel_launch(void* const* d_in, const int* in_sizes, int n_in,
                              void* d_out, int out_size, void* d_ws, size_t ws_size,
                              hipStream_t stream) {
    const float* start_x     = (const float*)d_in[0];
    const float* start_y     = (const float*)d_in[1];
    const float* start_yaw   = (const float*)d_in[2];
    const float* start_speed = (const float*)d_in[3];
    const float* accel       = (const float*)d_in[4];
    const float* steering    = (const float*)d_in[5];
    float*       out         = (float*)d_out;

    const int    B  = in_sizes[0];           // 65536
    const size_t nb = (size_t)B * N_STEPS;   // per-output-array stride

    dim3 grid(B / BLOCK), block(BLOCK);
    hipLaunchKernelGGL(bicycle_rollout_kernel, grid, block, 0, stream,
                       start_x, start_y, start_yaw, start_speed,
                       accel, steering, out, nb);
}